// ConvSP_28913719837051
// MI455X (gfx1250) — compile-verified
//
#include <hip/hip_runtime.h>

// ConvSP (SPH convolution), config: K=1 cell @ zero offset, 16->16 channels,
// B=2, N=32768, M=64 neighbors, h=0.25, poly6 kernel, no dis_norm/diffdata.
//
// out[b,n,o] = bias[o] + sum_c weight[o,c] * ( sum_m w(b,n,m) * data[b,idx,c] )
// w = 315/(64 pi h^9) * max(h^2 - d^2, 0)^3

typedef __attribute__((ext_vector_type(2))) float v2f;
typedef __attribute__((ext_vector_type(8))) float v8f;

#define B_ 2
#define N_ 32768
#define M_ 64
#define C_ 16
#define O_ 16
#define THREADS 256
#define WAVES (THREADS / 32)

__launch_bounds__(THREADS)
__global__ void convsp_wmma_kernel(const float* __restrict__ locs,
                                   const float* __restrict__ data,
                                   const int*   __restrict__ neighbors,
                                   const float* __restrict__ weight,
                                   const float* __restrict__ bias,
                                   float* __restrict__ out)
{
    // Per-wave staging slab: 32 points x 16 channels, stride 17 (17 coprime
    // with 64 banks -> conflict-free for both the write and the A-layout read).
    __shared__ float fshare[WAVES][32][C_ + 1];

    const int tid  = threadIdx.x;
    const int wave = tid >> 5;
    const int lane = tid & 31;
    const int gpt  = blockIdx.x * THREADS + tid;   // global point id in [0, B*N)
    const int b    = gpt >> 15;                    // gpt / N_   (N_ = 2^15)
    const int n    = gpt & (N_ - 1);

    const float* lbase = locs + (size_t)b * N_ * 3;
    const float* dbase = data + (size_t)b * N_ * C_;
    const int4*  nb4   = (const int4*)(neighbors + (size_t)gpt * M_);

    const float px = lbase[n * 3 + 0];
    const float py = lbase[n * 3 + 1];
    const float pz = lbase[n * 3 + 2];

    const float H    = 0.25f;
    const float H2   = H * H;
    const float coef = 315.0f / (64.0f * 3.14159265358979323846f *
                                 H * H * H * H * H * H * H * H * H);

    float f[C_];
#pragma unroll
    for (int c = 0; c < C_; ++c) f[c] = 0.0f;

    // ---- gather loop: 64 neighbors, indices streamed 4-at-a-time (b128) ----
    for (int m4 = 0; m4 < M_ / 4; ++m4) {
        int4 ids = nb4[m4];
        // prefetch the next 16B of this lane's index stream (global_prefetch_b8)
        if (m4 + 1 < M_ / 4) __builtin_prefetch(nb4 + m4 + 1, 0, 0);

        int idarr[4] = {ids.x, ids.y, ids.z, ids.w};
#pragma unroll
        for (int j = 0; j < 4; ++j) {
            int id = idarr[j];
            if (id < 0) continue;                  // invalid neighbor -> w = 0
            const float* nl = lbase + (size_t)id * 3;
            float dx = nl[0] - px;
            float dy = nl[1] - py;
            float dz = nl[2] - pz;
            float d2 = dx * dx + dy * dy + dz * dz;
            float t  = H2 - d2;
            if (t > 0.0f) {                        // ~6.5% of random pairs pass
                float w = coef * t * t * t;
                const float4* dp = (const float4*)(dbase + (size_t)id * C_);
#pragma unroll
                for (int q = 0; q < 4; ++q) {
                    float4 v = dp[q];
                    f[4 * q + 0] += w * v.x;
                    f[4 * q + 1] += w * v.y;
                    f[4 * q + 2] += w * v.z;
                    f[4 * q + 3] += w * v.w;
                }
            }
        }
    }

    // ---- stage f into LDS so it can be re-read in WMMA A layout ----
#pragma unroll
    for (int c = 0; c < C_; ++c) fshare[wave][lane][c] = f[c];
    __syncthreads();

    // ---- 16x16 channel mix via V_WMMA_F32_16X16X4_F32 (K=16 in 4 steps) ----
    // A (16x4 f32) layout: lanes 0-15 hold A[M=lane][K, K+1] in v0,v1;
    //                      lanes 16-31 hold A[M=lane-16][K+2, K+3].
    // B (4x16 f32) mirrors: lane&15 = column N, half selects K pair.
    // C/D (16x16 f32): VGPR r -> row r (lanes 0-15) / row r+8 (lanes 16-31),
    //                  column = lane&15.
    const int nloc = lane & 15;    // A row within tile / B,C,D column (= out ch)
    const int hi   = lane >> 4;    // half-wave selector

    v8f cinit;
    {
        float bv = bias[nloc];     // bias depends only on the output channel
#pragma unroll
        for (int r = 0; r < 8; ++r) cinit[r] = bv;
    }

#pragma unroll
    for (int t = 0; t < 2; ++t) {  // two 16-point tiles per wave
        v8f acc = cinit;
#pragma unroll
        for (int k = 0; k < 4; ++k) {
            const int kb = 4 * k + 2 * hi;
            v2f a, wv;
            a[0]  = fshare[wave][t * 16 + nloc][kb];
            a[1]  = fshare[wave][t * 16 + nloc][kb + 1];
            // B[K=c][N=o] = weight[o*C_ + c]
            wv[0] = weight[nloc * C_ + kb];
            wv[1] = weight[nloc * C_ + kb + 1];
            acc = __builtin_amdgcn_wmma_f32_16x16x4_f32(
                /*neg_a=*/false, a, /*neg_b=*/false, wv,
                /*c_mod=*/(short)0, acc, /*reuse_a=*/false, /*reuse_b=*/false);
        }
        const int tilebase = blockIdx.x * THREADS + wave * 32 + t * 16;
#pragma unroll
        for (int r = 0; r < 8; ++r) {
            const int mrow = r + 8 * hi;
            out[(size_t)(tilebase + mrow) * O_ + nloc] = acc[r];
        }
    }
}

extern "C" void kernel_launch(void* const* d_in, const int* in_sizes, int n_in,
                              void* d_out, int out_size, void* d_ws, size_t ws_size,
                              hipStream_t stream)
{
    const float* locs = (const float*)d_in[0];
    const float* data = (const float*)d_in[1];
    const int*   nbrs = (const int*)d_in[2];
    const float* wgt  = (const float*)d_in[3];
    const float* bias = (const float*)d_in[4];
    float* out = (float*)d_out;

    dim3 grid((B_ * N_) / THREADS);
    convsp_wmma_kernel<<<grid, THREADS, 0, stream>>>(locs, data, nbrs, wgt, bias, out);
}